// SmoothDiffusedSkinningField_66236985639649
// MI455X (gfx1250) — compile-verified
//
#include <hip/hip_runtime.h>
#include <hip/hip_bf16.h>
#include <stdint.h>

#define TPB 256
#define CCH 24
#define RES 96
#define R2  (RES * RES)        /* 9216   */
#define R3  (RES * RES * RES)  /* 884736 */

typedef __attribute__((address_space(1))) int gint_t;
typedef __attribute__((address_space(3))) int lint_t;

// gfx1250 async global->LDS copy (tracked with ASYNCcnt).
__device__ __forceinline__ void async_g2l_b32(const void* gsrc, void* ldst) {
    __builtin_amdgcn_global_load_async_to_lds_b32(
        (gint_t*)gsrc, (lint_t*)ldst, /*offset=*/0, /*cpol=*/0);
}

__global__ void __launch_bounds__(TPB)
sdsf_trilinear_kernel(const float* __restrict__ query,        // [B,3,N]
                      const int*   __restrict__ garment_idx,  // [B]
                      const int*   __restrict__ index_map,    // [S]
                      const float* __restrict__ fields,       // [S,C,96,96,96]
                      const float* __restrict__ bbox_extend,  // [S]
                      const float* __restrict__ bbox_center,  // [S,3]
                      float*       __restrict__ out,          // [B,C,N]
                      int N)
{
    __shared__ float sq[3 * TPB];

    const int tid = threadIdx.x;
    const int b   = blockIdx.y;
    const int n   = blockIdx.x * TPB + tid;
    if (n >= N) return;

    // Per-batch uniform scalars (b is uniform -> scalar loads).
    const int   sidx = index_map[garment_idx[b]];
    const float ext  = bbox_extend[sidx];
    const float cx   = bbox_center[sidx * 3 + 0];
    const float cy   = bbox_center[sidx * 3 + 1];
    const float cz   = bbox_center[sidx * 3 + 2];

    // Stage this block's coordinates through LDS with async copies (ASYNCcnt path).
    const float* q = query + (size_t)b * 3u * (size_t)N + (size_t)n;
    async_g2l_b32(q,                 &sq[tid]);
    async_g2l_b32(q + (size_t)N,     &sq[TPB + tid]);
    async_g2l_b32(q + 2 * (size_t)N, &sq[2 * TPB + tid]);
    asm volatile("s_wait_asynccnt 0" ::: "memory");

    const float vx = sq[tid];
    const float vy = sq[TPB + tid];
    const float vz = sq[2 * TPB + tid];

    // unnorm(c,96) with coords = (v-center)/ext*2 :
    //   i = clip(((c+1)*96-1)*0.5, 0, 95) = clip((v-center)*(96/ext) + 47.5, 0, 95)
    const float s96 = 96.0f / ext;
    float ix = fminf(fmaxf(fmaf(vx - cx, s96, 47.5f), 0.0f), 95.0f);
    float iy = fminf(fmaxf(fmaf(vy - cy, s96, 47.5f), 0.0f), 95.0f);
    float iz = fminf(fmaxf(fmaf(vz - cz, s96, 47.5f), 0.0f), 95.0f);

    // Border-clamp trick: clamp base cell to RES-2; frac absorbs the edge case
    // (ix==95 -> x0=94, fx=1 -> picks corner 95 exactly). Identical to reference.
    int x0 = (int)ix; x0 = (x0 > RES - 2) ? RES - 2 : x0;
    int y0 = (int)iy; y0 = (y0 > RES - 2) ? RES - 2 : y0;
    int z0 = (int)iz; z0 = (z0 > RES - 2) ? RES - 2 : z0;
    const float fx = ix - (float)x0;
    const float fy = iy - (float)y0;
    const float fz = iz - (float)z0;
    const float gx = 1.0f - fx, gy = 1.0f - fy, gz = 1.0f - fz;

    // 8 trilinear weights, computed once and reused over all 24 channels.
    const float wzy00 = gz * gy, wzy01 = gz * fy, wzy10 = fz * gy, wzy11 = fz * fy;
    const float w000 = wzy00 * gx, w001 = wzy00 * fx;
    const float w010 = wzy01 * gx, w011 = wzy01 * fx;
    const float w100 = wzy10 * gx, w101 = wzy10 * fx;
    const float w110 = wzy11 * gx, w111 = wzy11 * fx;

    // Uniform SGPR base pointer + 32-bit per-lane offset (GVS addressing);
    // corner deltas become 24-bit instruction offsets.
    const float* __restrict__ fb =
        fields + (size_t)sidx * (size_t)CCH * (size_t)R3;
    const uint32_t o0 = (uint32_t)((z0 * RES + y0) * RES + x0);

    float* __restrict__ op = out + ((size_t)b * CCH) * (size_t)N + (size_t)n;

#pragma unroll 4
    for (int c = 0; c < CCH; ++c) {
        const uint32_t oc = o0 + (uint32_t)c * (uint32_t)R3;
        const float g000 = fb[oc];
        const float g001 = fb[oc + 1];
        const float g010 = fb[oc + RES];
        const float g011 = fb[oc + RES + 1];
        const float g100 = fb[oc + R2];
        const float g101 = fb[oc + R2 + 1];
        const float g110 = fb[oc + R2 + RES];
        const float g111 = fb[oc + R2 + RES + 1];

        float acc = g000 * w000;
        acc = fmaf(g001, w001, acc);
        acc = fmaf(g010, w010, acc);
        acc = fmaf(g011, w011, acc);
        acc = fmaf(g100, w100, acc);
        acc = fmaf(g101, w101, acc);
        acc = fmaf(g110, w110, acc);
        acc = fmaf(g111, w111, acc);

        op[(size_t)c * (size_t)N] = acc;
    }
}

extern "C" void kernel_launch(void* const* d_in, const int* in_sizes, int n_in,
                              void* d_out, int out_size, void* d_ws, size_t ws_size,
                              hipStream_t stream) {
    const float* query       = (const float*)d_in[0];
    const int*   garment_idx = (const int*)  d_in[1];
    const int*   index_map   = (const int*)  d_in[2];
    const float* fields      = (const float*)d_in[3];
    const float* bbox_extend = (const float*)d_in[4];
    const float* bbox_center = (const float*)d_in[5];
    float*       out         = (float*)d_out;

    const int B = in_sizes[1];                 // 8
    const int N = in_sizes[0] / (3 * B);       // 65536

    dim3 grid((N + TPB - 1) / TPB, B);
    sdsf_trilinear_kernel<<<grid, TPB, 0, stream>>>(
        query, garment_idx, index_map, fields, bbox_extend, bbox_center, out, N);
}